// NPChangeTransitionPrior_45432164057759
// MI455X (gfx1250) — compile-verified
//
#include <hip/hip_runtime.h>
#include <math.h>

// Problem constants (from reference): B=32, T=1002, D=16, E=32, H=64, LAGS=2
#define B_    32
#define T_    1002
#define D_    16
#define E_    32
#define H_    64
#define LAGS_ 2
#define LEN_  1000
#define N_    (B_ * LEN_)      // 32000 samples
#define SLOPE 0.2f

typedef __attribute__((ext_vector_type(2))) float v2f;
typedef __attribute__((ext_vector_type(8))) float v8f;

// D = A(16x4) * B(4x16) + C, fp32 WMMA (CDNA5 V_WMMA_F32_16X16X4_F32)
__device__ __forceinline__ v8f wmma4(v2f a, v2f b, v8f c) {
  return __builtin_amdgcn_wmma_f32_16x16x4_f32(
      /*neg_a=*/false, a, /*neg_b=*/false, b,
      /*c_mod=*/(short)0, c, /*reuse_a=*/false, /*reuse_b=*/false);
}

// One hidden layer (64->64) for BOTH the forward chain (z = a_prev @ W^T + b,
// a = leaky(z)) and the Jacobian chain (u = t_prev @ W^T, t = g * u).
// A-side fragments come from per-wave LDS buffers pA (activations) and pT
// (t-chain); results are written back into the same buffers (all loads are
// hoisted before any store; same-wave LDS ops are in order).
__device__ __forceinline__ void layer64(const float* __restrict__ sW,   // [64][64] row-major (out x in)
                                        const float* __restrict__ sB,   // [64]
                                        float* __restrict__ pA,
                                        float* __restrict__ pT,
                                        int lane) {
  const int m    = lane & 15;   // A row = sample within tile
  const int half = lane >> 4;   // lane half selects K sub-pair / M sub-block

  // Load all A-side fragments (K = 0..63 in chunks of 4) for both chains.
  v2f af[16], tf[16];
#pragma unroll
  for (int kc = 0; kc < 16; ++kc) {
    const int k = kc * 4 + 2 * half;
    const float* pa = pA + m * H_ + k;
    const float* pt = pT + m * H_ + k;
    af[kc].x = pa[0]; af[kc].y = pa[1];
    tf[kc].x = pt[0]; tf[kc].y = pt[1];
  }

#pragma unroll
  for (int nt = 0; nt < 4; ++nt) {               // four 16-wide column tiles
    const int col = m + 16 * nt;                 // output neuron index (per lane)
    const float* wrow = sW + col * H_;           // W[col][*]; B[k][col] = W[col][k]
    v8f cz = {0.f, 0.f, 0.f, 0.f, 0.f, 0.f, 0.f, 0.f};
    v8f cu = {0.f, 0.f, 0.f, 0.f, 0.f, 0.f, 0.f, 0.f};
#pragma unroll
    for (int kc = 0; kc < 16; ++kc) {
      const int k = kc * 4 + 2 * half;
      v2f bf; bf.x = wrow[k]; bf.y = wrow[k + 1];
      cz = wmma4(af[kc], bf, cz);                // forward
      cu = wmma4(tf[kc], bf, cu);                // Jacobian (same B fragment)
    }
    const float bias = sB[col];
#pragma unroll
    for (int r = 0; r < 8; ++r) {                // accumulator rows
      const int   row = r + 8 * half;            // sample row 0..15
      const float z   = cz[r] + bias;
      const float g   = (z >= 0.f) ? 1.f : SLOPE;
      pA[row * H_ + col] = z * g;                // a = leaky(z)
      pT[row * H_ + col] = g * cu[r];            // t = g * (W @ t_prev)
    }
  }
}

__global__ __launch_bounds__(64) void mlp_flow_kernel(
    const float* __restrict__ x,    // [B, T, D]
    const float* __restrict__ emb,  // [B, T, E]
    const float* __restrict__ W0,   // [D, H, E+1]
    const float* __restrict__ b0,   // [D, H]
    const float* __restrict__ W1,   // [D, H, H]
    const float* __restrict__ b1,
    const float* __restrict__ W2,
    const float* __restrict__ b2,
    const float* __restrict__ Wo,   // [D, 1, H]
    const float* __restrict__ bo,   // [D, 1]
    float* __restrict__ out_res,    // [N, D]
    float* __restrict__ ws_log)     // [N, D] scratch: log|dJ| terms
{
  __shared__ float sW0[H_ * (E_ + 1)];
  __shared__ float sW1[H_ * H_];
  __shared__ float sW2[H_ * H_];
  __shared__ float sB0[H_], sB1[H_], sB2[H_], sWo[H_];
  __shared__ float sA[2][16 * H_];   // per-wave activation buffer
  __shared__ float sT[2][16 * H_];   // per-wave Jacobian-chain buffer

  const int tid  = threadIdx.x;
  const int wave = tid >> 5;
  const int lane = tid & 31;
  const int d    = blockIdx.x & (D_ - 1);     // latent dim
  const int pair = blockIdx.x >> 4;           // 0..999
  const int tile = pair * 2 + wave;           // 16-sample tile, 0..1999

  // ---- stage weights for latent dim d into LDS (cooperative) ----
  {
    const float* gW0 = W0 + d * H_ * (E_ + 1);
    for (int i = tid; i < H_ * (E_ + 1); i += 64) sW0[i] = gW0[i];
    const float* gW1 = W1 + d * H_ * H_;
    const float* gW2 = W2 + d * H_ * H_;
    for (int i = tid; i < H_ * H_; i += 64) { sW1[i] = gW1[i]; sW2[i] = gW2[i]; }
    if (tid < H_) {
      sB0[tid] = b0[d * H_ + tid];
      sB1[tid] = b1[d * H_ + tid];
      sB2[tid] = b2[d * H_ + tid];
      sWo[tid] = Wo[d * H_ + tid];
    }
  }
  __syncthreads();

  float* pA = sA[wave];
  float* pT = sT[wave];

  const int m    = lane & 15;
  const int half = lane >> 4;
  const int n    = tile * 16 + m;             // global sample index for this lane's row
  const int bb   = n / LEN_;
  const int tt   = n % LEN_;
  const float* embRow = emb + ((size_t)bb * T_ + (tt + LAGS_)) * E_;
  const float  xval   = x[((size_t)bb * T_ + (tt + LAGS_)) * D_ + d];

  // ---- layer 0: z0 = [emb | x] @ W0^T + b0  (K = 33 = 8 chunks + x chunk) ----
  {
    v2f af0[8];
#pragma unroll
    for (int kc = 0; kc < 8; ++kc) {
      const int e = kc * 4 + 2 * half;
      af0[kc].x = embRow[e];
      af0[kc].y = embRow[e + 1];
    }
    v2f ax; ax.x = half ? 0.f : xval; ax.y = 0.f;   // zero-padded K=4 chunk for column 32

#pragma unroll
    for (int nt = 0; nt < 4; ++nt) {
      const int col = m + 16 * nt;
      const float* wrow = sW0 + col * (E_ + 1);
      v8f cz = {0.f, 0.f, 0.f, 0.f, 0.f, 0.f, 0.f, 0.f};
#pragma unroll
      for (int kc = 0; kc < 8; ++kc) {
        const int e = kc * 4 + 2 * half;
        v2f bf; bf.x = wrow[e]; bf.y = wrow[e + 1];
        cz = wmma4(af0[kc], bf, cz);
      }
      {
        v2f bf; bf.x = half ? 0.f : wrow[E_]; bf.y = 0.f;
        cz = wmma4(ax, bf, cz);
      }
      const float bias = sB0[col];
      const float w0c  = wrow[E_];                  // W0[d][col][32] (dz0/dx column)
#pragma unroll
      for (int r = 0; r < 8; ++r) {
        const int   row = r + 8 * half;
        const float z   = cz[r] + bias;
        const float g   = (z >= 0.f) ? 1.f : SLOPE;
        pA[row * H_ + col] = z * g;                 // a0
        pT[row * H_ + col] = g * w0c;               // t0 = g0 * W0[:, -1]
      }
    }
  }

  // ---- layers 1 and 2 (forward + Jacobian chain fused) ----
  layer64(sW1, sB1, pA, pT, lane);
  layer64(sW2, sB2, pA, pT, lane);

  // ---- output heads: res = Wo @ a2 + bo ; dJ = Wo @ t2 ----
  // half 0 lanes reduce activations -> residual, half 1 lanes reduce t -> log|dJ|
  {
    const float* buf = half ? pT : pA;
    float acc = 0.f;
#pragma unroll
    for (int h = 0; h < H_; ++h) acc = fmaf(buf[m * H_ + h], sWo[h], acc);
    if (half == 0) {
      out_res[(size_t)n * D_ + d] = acc + bo[d];
    } else {
      ws_log[(size_t)n * D_ + d] = logf(fabsf(acc));
    }
  }
}

__global__ __launch_bounds__(256) void reduce_logdet_kernel(
    const float* __restrict__ ws_log,   // [N, D]
    float* __restrict__ out_ld)         // [N]
{
  const int n = blockIdx.x * blockDim.x + threadIdx.x;
  if (n < N_) {
    float s = 0.f;
#pragma unroll
    for (int dd = 0; dd < D_; ++dd) s += ws_log[(size_t)n * D_ + dd];
    out_ld[n] = s;
  }
}

extern "C" void kernel_launch(void* const* d_in, const int* in_sizes, int n_in,
                              void* d_out, int out_size, void* d_ws, size_t ws_size,
                              hipStream_t stream) {
  const float* x   = (const float*)d_in[0];
  const float* emb = (const float*)d_in[1];
  const float* W0  = (const float*)d_in[2];
  const float* b0  = (const float*)d_in[3];
  const float* W1  = (const float*)d_in[4];
  const float* b1  = (const float*)d_in[5];
  const float* W2  = (const float*)d_in[6];
  const float* b2  = (const float*)d_in[7];
  const float* Wo  = (const float*)d_in[8];
  const float* bo  = (const float*)d_in[9];

  float* out_res = (float*)d_out;                 // residuals [B, LEN, D] flat
  float* out_ld  = out_res + (size_t)N_ * D_;     // log_abs_det [B, LEN] flat
  float* ws_log  = (float*)d_ws;                  // [N, D] log terms (2 MB)

  // 16 latent dims x 1000 tile-pairs; each block: 2 waves, each owning one
  // 16-sample tile for dim d.
  const int grid1 = D_ * (N_ / 16 / 2);           // 16000 blocks
  mlp_flow_kernel<<<grid1, 64, 0, stream>>>(x, emb, W0, b0, W1, b1, W2, b2,
                                            Wo, bo, out_res, ws_log);

  reduce_logdet_kernel<<<(N_ + 255) / 256, 256, 0, stream>>>(ws_log, out_ld);
}